// MyModel_87522843561242
// MI455X (gfx1250) — compile-verified
//
#include <hip/hip_runtime.h>

// Problem constants (from reference): B=2048, T=2048, F=8, U=3, 3U=9.
#define B_DIM  2048
#define T_DIM  2048
#define F_DIM  8
#define GU     9      // 3*U gate columns
#define REC    12     // record per (b,t): 9 gates + mask + 2 pad -> 48B (16B aligned)
#define ROWSTR 20     // LDS row stride in floats (breaks 64-bank conflict between half-waves)

typedef float v2f __attribute__((ext_vector_type(2)));
typedef float v8f __attribute__((ext_vector_type(8)));

#if defined(__HIP_DEVICE_COMPILE__)
// gfx1250 hardware tanh (TRANS op). v_nop after per ISA 7.4 TRANS hazard rule.
__device__ __forceinline__ float tanh_fast(float x) {
  float r;
  asm volatile("v_tanh_f32 %0, %1\n\tv_nop" : "=v"(r) : "v"(x));
  return r;
}
__device__ __forceinline__ float sig_fast(float x) {
  return fmaf(0.5f, tanh_fast(0.5f * x), 0.5f);
}
#endif

// ---------------------------------------------------------------------------
// Phase 1: record[r = b*T+t] = { x[r,:] @ kernel + bias_i (9), mask, 0, 0 }
// Each wave: 32 consecutive rows (2 tiles of 16), 4x V_WMMA_F32_16X16X4_F32,
// LDS transpose, then 3 fully-coalesced 512B global_store_b128.
// ---------------------------------------------------------------------------
__global__ __launch_bounds__(256) void gru_proj_wmma(
    const float* __restrict__ x, const float* __restrict__ kern,
    const float* __restrict__ bias_i, float* __restrict__ xw)
{
#if defined(__HIP_DEVICE_COMPILE__)
  __shared__ float lds[8 * 32 * ROWSTR];          // 20KB, per-wave regions
  const int lane = threadIdx.x & 31;
  const int n    = lane & 15;                     // output column / row-in-tile
  const int hi   = lane >> 4;                     // half-wave: K pair select
  const int wv   = threadIdx.x >> 5;
  float* wl = lds + wv * (32 * ROWSTR);

  const long grp = (long)blockIdx.x * 8 + wv;
  const long r0  = grp << 5;                      // first of 32 rows (b uniform: T%32==0)

  // A-matrices: 32-bit 16x4 layout -> lane holds row M=n, K = 2*hi + {0,1}.
  const float2* xpA = (const float2*)(x + (r0 + n) * F_DIM);
  const float2* xpB = (const float2*)(x + (r0 + 16 + n) * F_DIM);
  const float2 aA01 = xpA[hi], aA45 = xpA[2 + hi];
  const float2 aB01 = xpB[hi], aB45 = xpB[2 + hi];

  // Row masks: any(x[row][0..7] != 0); half-waves each see 4 of 8 elems.
  float mpA = (aA01.x != 0.f || aA01.y != 0.f || aA45.x != 0.f || aA45.y != 0.f) ? 1.f : 0.f;
  float mpB = (aB01.x != 0.f || aB01.y != 0.f || aB45.x != 0.f || aB45.y != 0.f) ? 1.f : 0.f;
  const float moA = __shfl_xor(mpA, 16, 32);
  const float moB = __shfl_xor(mpB, 16, 32);
  const float mfA = (mpA != 0.f || moA != 0.f) ? 1.f : 0.f;   // mask of row r0+n
  const float mfB = (mpB != 0.f || moB != 0.f) ? 1.f : 0.f;   // mask of row r0+16+n

  // B-matrix (kernel 8x9 zero-padded to two 4x16 tiles) and C = bias_i.
  const int kb = 2 * hi;
  float b10 = 0.f, b11 = 0.f, b20 = 0.f, b21 = 0.f, cinit = 0.f;
  if (n < GU) {
    b10 = kern[(kb + 0) * GU + n];
    b11 = kern[(kb + 1) * GU + n];
    b20 = kern[(kb + 4) * GU + n];
    b21 = kern[(kb + 5) * GU + n];
    cinit = bias_i[n];
  }
  v8f cA, cB;
#pragma unroll
  for (int i = 0; i < 8; ++i) { cA[i] = cinit; cB[i] = cinit; }

  const v2f bm1 = {b10, b11}, bm2 = {b20, b21};
  const v2f a1A = {aA01.x, aA01.y}, a2A = {aA45.x, aA45.y};
  const v2f a1B = {aB01.x, aB01.y}, a2B = {aB45.x, aB45.y};
  cA = __builtin_amdgcn_wmma_f32_16x16x4_f32(false, a1A, false, bm1, (short)0, cA, false, false);
  cA = __builtin_amdgcn_wmma_f32_16x16x4_f32(false, a2A, false, bm2, (short)0, cA, false, false);
  cB = __builtin_amdgcn_wmma_f32_16x16x4_f32(false, a1B, false, bm1, (short)0, cB, false, false);
  cB = __builtin_amdgcn_wmma_f32_16x16x4_f32(false, a2B, false, bm2, (short)0, cB, false, false);

  // Transpose through LDS. D layout: VGPR i = row (hi*8 + i), col n.
  // Affine base + constant offsets -> ds_store immediate-offset forms.
  float* wrow = wl + (hi * 8) * ROWSTR + n;       // single base VGPR
#pragma unroll
  for (int i = 0; i < 8; ++i) {
    wrow[i * ROWSTR]                = cA[i];      // tile A: rows hi*8 + i
    wrow[i * ROWSTR + 16 * ROWSTR]  = cB[i];      // tile B: +1280B immediate
  }
  // Mask into col 9 (after the c-writes: same-wave DS ops are in-order), pads zeroed.
  wl[(hi * 16 + n) * ROWSTR + 9] = hi ? mfB : mfA;
  *(float2*)(wl + lane * ROWSTR + 10) = make_float2(0.f, 0.f);  // row=lane, cols 10-11

  asm volatile("" ::: "memory");   // keep LDS stores before the re-loads

  // Read back row-major and store 96 float4 chunks = 3 fully-coalesced b128 stores.
  float* gbase = xw + r0 * REC;    // 32*REC = 384 floats contiguous
#pragma unroll
  for (int k = 0; k < 3; ++k) {
    const int ci = lane + 32 * k;  // chunk: row j = ci/3, quad q = ci%3
    const int j  = ci / 3;
    const int q  = ci - 3 * j;
    const float4 v = *(const float4*)(wl + j * ROWSTR + q * 4);
    *(float4*)(gbase + ci * 4) = v;
  }
#endif  // __HIP_DEVICE_COMPILE__
}

// ---------------------------------------------------------------------------
// Phase 2: sequential GRU scan, one lane per batch row, dense head fused.
// Each lane streams its own contiguous 96KB record region (layout (b,t,REC)).
// ---------------------------------------------------------------------------
__global__ __launch_bounds__(256) void gru_scan(
    const float* __restrict__ xw, const float* __restrict__ rk,
    const float* __restrict__ bias_r, const float* __restrict__ dw,
    const float* __restrict__ dbp, float* __restrict__ out)
{
#if defined(__HIP_DEVICE_COMPILE__)
  const int b = blockIdx.x * blockDim.x + threadIdx.x;
  if (b >= B_DIM) return;

  float R[3][GU];
#pragma unroll
  for (int v = 0; v < 3; ++v)
#pragma unroll
    for (int u = 0; u < GU; ++u) R[v][u] = rk[v * GU + u];
  float br[GU];
#pragma unroll
  for (int u = 0; u < GU; ++u) br[u] = bias_r[u];
  const float w0 = dw[0], w1 = dw[1], w2 = dw[2];
  const float db = dbp[0];

  float h0 = 0.f, h1 = 0.f, h2 = 0.f;
  const float* rec = xw + (long)b * T_DIM * REC;   // per-lane sequential stream
  float*       yp  = out + (long)b * T_DIM;
  float        ybuf[4];

#pragma unroll 4
  for (int t = 0; t < T_DIM; ++t) {
    const float4* rp = (const float4*)(rec + (long)t * REC);
    const float4 g0 = rp[0];   // xz0 xz1 xz2 xr0
    const float4 g1 = rp[1];   // xr1 xr2 xh0 xh1
    const float4 g2 = rp[2];   // xh2 mask 0 0

    // gfx1250 global_prefetch_b8: once per unrolled group, 16 records (768B) ahead.
    if ((t & 3) == 0) {
      const int tp = (t + 16 < T_DIM) ? (t + 16) : (T_DIM - 1);
      __builtin_prefetch(rec + (long)tp * REC, 0, 0);
    }

    float hw[GU];
#pragma unroll
    for (int u = 0; u < GU; ++u)
      hw[u] = fmaf(h2, R[2][u], fmaf(h1, R[1][u], fmaf(h0, R[0][u], br[u])));

    const float z0 = sig_fast(g0.x + hw[0]);
    const float z1 = sig_fast(g0.y + hw[1]);
    const float z2 = sig_fast(g0.z + hw[2]);
    const float r0 = sig_fast(g0.w + hw[3]);
    const float r1 = sig_fast(g1.x + hw[4]);
    const float r2 = sig_fast(g1.y + hw[5]);
    const float hc0 = tanh_fast(fmaf(r0, hw[6], g1.z));
    const float hc1 = tanh_fast(fmaf(r1, hw[7], g1.w));
    const float hc2 = tanh_fast(fmaf(r2, hw[8], g2.x));

    // h_new_full = z*h + (1-z)*hc = hc + z*(h-hc)
    const float hn0 = fmaf(z0, h0 - hc0, hc0);
    const float hn1 = fmaf(z1, h1 - hc1, hc1);
    const float hn2 = fmaf(z2, h2 - hc2, hc2);

    const bool m = (g2.y != 0.f);
    const float y = m ? fmaf(hn0, w0, fmaf(hn1, w1, fmaf(hn2, w2, db))) : db;
    h0 = m ? hn0 : h0;
    h1 = m ? hn1 : h1;
    h2 = m ? hn2 : h2;

    ybuf[t & 3] = y;
    if ((t & 3) == 3)
      *(float4*)(yp + t - 3) = make_float4(ybuf[0], ybuf[1], ybuf[2], ybuf[3]);
  }
#endif  // __HIP_DEVICE_COMPILE__
}

// ---------------------------------------------------------------------------
extern "C" void kernel_launch(void* const* d_in, const int* in_sizes, int n_in,
                              void* d_out, int out_size, void* d_ws, size_t ws_size,
                              hipStream_t stream) {
  const float* x    = (const float*)d_in[0];  // (B,T,F)
  const float* kern = (const float*)d_in[1];  // (F,3U)
  const float* rk   = (const float*)d_in[2];  // (U,3U)
  const float* bi   = (const float*)d_in[3];  // (3U,)
  const float* brr  = (const float*)d_in[4];  // (3U,)
  const float* dw   = (const float*)d_in[5];  // (U,1)
  const float* db   = (const float*)d_in[6];  // (1,)
  float* out = (float*)d_out;                 // (B,T,1) -> b*T + t
  float* xw  = (float*)d_ws;                  // needs B*T*REC*4 = 192 MiB

  (void)in_sizes; (void)n_in; (void)out_size; (void)ws_size;

  // Phase 1: 32 rows per wave, 8 waves per block -> 256 rows per block.
  const long groups = (long)B_DIM * T_DIM / 32;   // 131072
  const int  blocks = (int)(groups / 8);          // 16384
  gru_proj_wmma<<<blocks, 256, 0, stream>>>(x, kern, bi, xw);

  // Phase 2: one lane per sequence.
  gru_scan<<<(B_DIM + 255) / 256, 256, 0, stream>>>(xw, rk, brr, dw, db, out);
}